// SpatialAttentionBlock_6571299962982
// MI455X (gfx1250) — compile-verified
//
#include <hip/hip_runtime.h>
#include <math.h>

#define BATCH 32
#define CH    256
#define HH    64
#define WW    64
#define HW    (HH * WW)       // 4096
#define CHW   (CH * HW)       // 1048576
#define KS    7
#define PAD   3
#define KFLAT 98              // 2*7*7

typedef float v2f __attribute__((ext_vector_type(2)));
typedef float v4f __attribute__((ext_vector_type(4)));
typedef float v8f __attribute__((ext_vector_type(8)));

__device__ inline v4f v4max(v4f a, v4f b) {
    v4f r;
    r.x = fmaxf(a.x, b.x); r.y = fmaxf(a.y, b.y);
    r.z = fmaxf(a.z, b.z); r.w = fmaxf(a.w, b.w);
    return r;
}

// ---------------------------------------------------------------------------
// Kernel 1: channel max + mean. Block = 256 threads = 16 channel-groups x
// 16 float4 pixel-columns, covering 64 consecutive pixels (one image row).
// Grid = B*HW/64 = 2048 blocks. Fully coalesced b128 streaming of x (keeps
// x resident in the 192MB L2 for kernel 3).
// ---------------------------------------------------------------------------
__global__ __launch_bounds__(256) void sab_reduce_kernel(
    const float* __restrict__ x, float* __restrict__ maxp, float* __restrict__ avgp) {
    __shared__ v4f s_sum[16][16];
    __shared__ v4f s_max[16][16];

    const int t   = threadIdx.x;
    const int col = t & 15;   // float4 column within 64-pixel tile
    const int cg  = t >> 4;   // channel group (16 channels each)

    const int pix0 = blockIdx.x * 64 + col * 4;   // flat pixel over B*HW
    const int b    = pix0 >> 12;                  // /4096
    const int s    = pix0 & 4095;

    const float* xb = x + b * CHW + cg * 16 * HW + s;

    v4f vsum = {0.f, 0.f, 0.f, 0.f};
    v4f vmax = {-3.402823466e38f, -3.402823466e38f, -3.402823466e38f, -3.402823466e38f};

#pragma unroll
    for (int c = 0; c < 16; ++c) {
        const v4f v = *(const v4f*)(xb + c * HW);
        vsum += v;
        vmax = v4max(vmax, v);
    }

    s_sum[cg][col] = vsum;
    s_max[cg][col] = vmax;
    __syncthreads();

#pragma unroll
    for (int off = 8; off > 0; off >>= 1) {
        if (cg < off) {
            s_sum[cg][col] += s_sum[cg + off][col];
            s_max[cg][col]  = v4max(s_max[cg][col], s_max[cg + off][col]);
        }
        __syncthreads();
    }

    if (cg == 0) {
        const v4f m  = s_max[0][col];
        const v4f av = s_sum[0][col] * (1.0f / 256.0f);
        *(v4f*)(maxp + pix0) = m;
        *(v4f*)(avgp + pix0) = av;
    }
}

// ---------------------------------------------------------------------------
// Kernel 2: 7x7 conv (2->1 ch) + sigmoid via V_WMMA_F32_16X16X4_F32.
// One wave computes 16 output pixels (one N-tile). A = flattened 98 weights
// (padded to 100), identical in all 16 rows; B = 4x16 patch values.
// 25 WMMAs accumulate K=100. All D rows identical -> acc[0] is the result
// for pixel (lane & 15).
// Lane layouts (ISA 7.12.2, f32):
//   A 16x4: lane L (0-15): v0=A[L,0], v1=A[L,1]; lane L+16: v0=A[L,2], v1=A[L,3]
//   B 4x16: lane L (0-15): v0=B[0,L], v1=B[1,L]; lane L+16: v0=B[2,L], v1=B[3,L]
// ---------------------------------------------------------------------------
__device__ inline float sab_patch(int k, int b, int y, int px,
                                  const float* __restrict__ maxp,
                                  const float* __restrict__ avgp) {
    if (k >= KFLAT) return 0.0f;
    const int ci = k / 49;
    const int r  = k - ci * 49;
    const int ky = r / 7;
    const int kx = r - ky * 7;
    const int yy = y + ky - PAD;
    const int xx = px + kx - PAD;
    if (yy < 0 || yy >= HH || xx < 0 || xx >= WW) return 0.0f;
    const float* p = ci ? avgp : maxp;
    return p[b * HW + yy * WW + xx];
}

__global__ __launch_bounds__(256) void sab_conv_wmma_kernel(
    const float* __restrict__ w, const float* __restrict__ maxp,
    const float* __restrict__ avgp, float* __restrict__ scalep) {
    const int wave = (blockIdx.x * blockDim.x + threadIdx.x) >> 5;  // 0..8191
    const int lane = threadIdx.x & 31;
    const int n    = lane & 15;
    const int half = lane >> 4;

    const int b   = wave >> 8;        // tile / 256
    const int rem = wave & 255;
    const int y   = rem >> 2;         // 4 tiles per 64-wide row
    const int x0  = (rem & 3) << 4;
    const int px  = x0 + n;

    v8f acc = {};
#pragma unroll 1
    for (int kb = 0; kb < 100; kb += 4) {
        const int k0 = kb + half * 2;
        const int k1 = k0 + 1;
        v2f a, bb;
        a.x  = (k0 < KFLAT) ? w[k0] : 0.0f;
        a.y  = (k1 < KFLAT) ? w[k1] : 0.0f;
        bb.x = sab_patch(k0, b, y, px, maxp, avgp);
        bb.y = sab_patch(k1, b, y, px, maxp, avgp);
        acc  = __builtin_amdgcn_wmma_f32_16x16x4_f32(
                   /*neg_a=*/false, a, /*neg_b=*/false, bb,
                   /*c_mod=*/(short)0, acc, /*reuse_a=*/false, /*reuse_b=*/false);
    }

    const float conv = acc[0];
    const float sc   = 1.0f / (1.0f + expf(-conv));
    if (half == 0) {
        scalep[b * HW + y * WW + px] = sc;
    }
}

// ---------------------------------------------------------------------------
// Kernel 3: out = x * scale (broadcast over channels). b128 per thread.
// Non-temporal load of x (last use) and non-temporal store of out so the
// output stream doesn't evict L2-resident data; the 0.5MB scale plane is
// re-read 256x and stays cached.
// ---------------------------------------------------------------------------
__global__ __launch_bounds__(256) void sab_scale_kernel(
    const float* __restrict__ x, const float* __restrict__ scalep,
    float* __restrict__ out) {
    const int i4 = blockIdx.x * blockDim.x + threadIdx.x;  // float4 index, < 8388608
    const int e  = i4 << 2;                                // element index
    const int b  = e >> 20;                                // / CHW
    const int s  = e & 4095;                               // spatial within plane

    const v4f xv = __builtin_nontemporal_load((const v4f*)x + i4);
    const v4f sv = *(const v4f*)(scalep + b * HW + s);
    __builtin_nontemporal_store(xv * sv, (v4f*)out + i4);
}

extern "C" void kernel_launch(void* const* d_in, const int* in_sizes, int n_in,
                              void* d_out, int out_size, void* d_ws, size_t ws_size,
                              hipStream_t stream) {
    const float* x = (const float*)d_in[0];   // [32,256,64,64]
    const float* w = (const float*)d_in[1];   // [1,2,7,7] -> 98 floats flat
    float* out     = (float*)d_out;           // [32,256,64,64]

    float* maxp   = (float*)d_ws;             // [32,64,64]
    float* avgp   = maxp + BATCH * HW;        // [32,64,64]
    float* scalep = avgp + BATCH * HW;        // [32,64,64]

    // Pass 1: channel max/mean. 2048 blocks x 256 threads.
    sab_reduce_kernel<<<(BATCH * HW) / 64, 256, 0, stream>>>(x, maxp, avgp);

    // Pass 2: WMMA conv + sigmoid. 8192 waves = 1024 blocks x 8 waves.
    sab_conv_wmma_kernel<<<(BATCH * HW / 16) / 8, 256, 0, stream>>>(w, maxp, avgp, scalep);

    // Pass 3: broadcast multiply. 8388608 float4 threads.
    sab_scale_kernel<<<(BATCH * CHW / 4) / 256, 256, 0, stream>>>(x, scalep, out);
}